// RNN_34943853920814
// MI455X (gfx1250) — compile-verified
//
#include <hip/hip_runtime.h>
#include <hip/hip_bf16.h>

// ---------------------------------------------------------------------------
// RNN forward for MI455X (gfx1250, wave32, WMMA).
//   L=512, N=64, I=H=1024.
//   Phase 1: XW = x @ w_input              (big GEMM, fp32 WMMA)  -> d_out
//   Phase 2: h_t = tanh(h_{t-1}@Wh + XW_t) (persistent kernel, grid barrier,
//            w_hidden strip staged in LDS ONCE for all 512 steps,
//            overwrites XW_t with h_t in place)
//   Phase 3: copy h_last, then outputs = H_ALL @ w_output IN PLACE over d_out
//            (per-block row ownership + LDS staging makes in-place safe).
// All matrix math uses V_WMMA_F32_16X16X4_F32 (native fp32 matrix pipe:
// exact-precision match to the fp32 reference through the 512-step tanh
// recurrence, which bf16 would not survive).
// ---------------------------------------------------------------------------

typedef __attribute__((ext_vector_type(2))) float v2f;
typedef __attribute__((ext_vector_type(8))) float v8f;

#define LSTEPS 512
#define NB     64
#define HD     1024
#define MTOT   (LSTEPS * NB)   // 32768 rows for the big GEMMs
#define NH     (NB * HD)       // 65536 elements per timestep slab

// fp32 WMMA tile helper: one wave, one 16x16 tile, rank-4 update.
// Layout (ISA 7.12.2, 32-bit operands):
//   A (16x4): lane L -> row m=L%16, float2 {A[m][k+2*(L/16)], A[m][k+1+2*(L/16)]}
//   B (4x16): lane L -> col n=L%16, float2 {B[kk][n], B[kk+1][n]}, kk=k+2*(L/16)
//   C (16x16): lane L, vgpr r -> C[8*(L/16)+r][L%16]
static __device__ __forceinline__ v8f wmma_f32(v2f a, v2f b, v8f c) {
    return __builtin_amdgcn_wmma_f32_16x16x4_f32(false, a, false, b, (short)0, c,
                                                 false, false);
}

// ---------------------------------------------------------------------------
// Phase 1: C[32768,1024] = A[32768,1024] @ B[1024,1024]
// block = 128 threads (4 waves); each wave computes a 16x64 strip (4 tiles,
// A operand reused 4x). grid = (H/256, MTOT/16) = (4, 2048).
// ---------------------------------------------------------------------------
__global__ __launch_bounds__(128) void xw_gemm(const float* __restrict__ A,
                                               const float* __restrict__ B,
                                               float* __restrict__ C) {
    const int lane  = threadIdx.x & 31;
    const int wave  = threadIdx.x >> 5;
    const int m0    = blockIdx.y * 16;
    const int n0    = blockIdx.x * 256 + wave * 64;
    const int col   = lane & 15;
    const int koff  = 2 * (lane >> 4);            // 0 or 2
    const float* Arow = A + (m0 + col) * HD;      // col doubles as row-in-tile

    v8f acc0 = {}, acc1 = {}, acc2 = {}, acc3 = {};
#pragma unroll 4
    for (int k = 0; k < HD; k += 4) {
        const int kk = k + koff;
        v2f a = *(const v2f*)(Arow + kk);
        const float* Bp = B + kk * HD + n0 + col;
        v2f b0 = {Bp[0],      Bp[HD]};
        v2f b1 = {Bp[16],     Bp[HD + 16]};
        v2f b2 = {Bp[32],     Bp[HD + 32]};
        v2f b3 = {Bp[48],     Bp[HD + 48]};
        acc0 = wmma_f32(a, b0, acc0);
        acc1 = wmma_f32(a, b1, acc1);
        acc2 = wmma_f32(a, b2, acc2);
        acc3 = wmma_f32(a, b3, acc3);
    }
    const int rbase = m0 + 8 * (lane >> 4);
#pragma unroll
    for (int r = 0; r < 8; ++r) {
        float* Cp = C + (rbase + r) * HD + n0 + col;   // coalesced across lanes
        Cp[0] = acc0[r]; Cp[16] = acc1[r]; Cp[32] = acc2[r]; Cp[48] = acc3[r];
    }
}

// ---------------------------------------------------------------------------
// Phase 2: persistent recurrence. 64 blocks x 4 waves = 256 waves = one wave
// per 16x16 tile of h_t. All 4 waves of block b share the SAME w_hidden
// column strip (16 cols x 1024 rows = 64KB): staged into LDS once, reused for
// all 512 steps. Pair-interleaved layout Bs[(k/2)*32 + c*2 + (k&1)] makes the
// B operand a single conflict-free ds_load_b64 per wmma.
// HX holds XW on entry; slab t is overwritten with h_t.
// Grid-wide sense barrier between steps (bar[0]=count, bar[1]=generation).
// ---------------------------------------------------------------------------
__global__ __launch_bounds__(128) void rnn_steps(const float* __restrict__ Wh,
                                                 float* __restrict__ HX,
                                                 unsigned* __restrict__ bar) {
    __shared__ float Bs[HD * 16];          // 64KB (WGP has 320KB LDS)
    const int lane  = threadIdx.x & 31;
    const int wave  = threadIdx.x >> 5;
    const int n0    = blockIdx.x * 16;     // 64 blocks cover H
    const int m0    = wave * 16;           // 4 waves cover N=64
    const int col   = lane & 15;
    const int rbase = m0 + 8 * (lane >> 4);

    // one-time stage of the block's w_hidden strip, pair-interleaved over k
    for (int i = threadIdx.x; i < HD * 16; i += 128) {
        const int k = i >> 4;              // 0..1023
        const int c = i & 15;              // 0..15
        Bs[(k >> 1) * 32 + c * 2 + (k & 1)] = Wh[k * HD + n0 + c];
    }
    __syncthreads();

    const float* BsLane = &Bs[(lane >> 4) * 32 + col * 2];  // + (k/2)*32*(pairs)

    for (int t = 0; t < LSTEPS; ++t) {
        v8f acc = {};
        if (t > 0) {                       // h_0 = 0 -> skip the matmul
            const float* Arow = HX + (t - 1) * NH + (m0 + col) * HD;
#pragma unroll 4
            for (int k = 0; k < HD; k += 4) {
                const int kk = k + 2 * (lane >> 4);
                v2f a = *(const v2f*)(Arow + kk);
                v2f b = *(const v2f*)(BsLane + (k >> 1) * 32);
                acc = wmma_f32(a, b, acc);
            }
        }
        float* X = HX + t * NH;
#pragma unroll
        for (int r = 0; r < 8; ++r) {
            float* p = X + (rbase + r) * HD + n0 + col;
            *p = tanhf(acc[r] + *p);       // fused bias-add + activation
        }
        // ---- device-scope grid barrier ----
        __threadfence();                   // order h_t stores before signal
        __syncthreads();
        if (threadIdx.x == 0) {
            unsigned g = __hip_atomic_load(bar + 1, __ATOMIC_RELAXED,
                                           __HIP_MEMORY_SCOPE_AGENT);
            unsigned arrived = __hip_atomic_fetch_add(bar, 1u, __ATOMIC_ACQ_REL,
                                                      __HIP_MEMORY_SCOPE_AGENT);
            if (arrived == 63u) {          // last block: reset then bump phase
                __hip_atomic_store(bar, 0u, __ATOMIC_RELAXED,
                                   __HIP_MEMORY_SCOPE_AGENT);
                __hip_atomic_store(bar + 1, g + 1u, __ATOMIC_RELEASE,
                                   __HIP_MEMORY_SCOPE_AGENT);
            } else {
                while (__hip_atomic_load(bar + 1, __ATOMIC_ACQUIRE,
                                         __HIP_MEMORY_SCOPE_AGENT) == g) {
                    __builtin_amdgcn_s_sleep(2);
                }
            }
        }
        __syncthreads();
    }
}

// ---------------------------------------------------------------------------
// Phase 3: in-place C[32768,1024] = C @ B. Each block owns 16 rows: stage them
// into LDS (row stride 1026 words: conflict-free float2 A-operand reads, 8B
// aligned), sync, then 8 waves x 8 n-tiles compute and overwrite the rows.
// ---------------------------------------------------------------------------
#define AS_STRIDE 1026
__global__ __launch_bounds__(256) void out_gemm_inplace(float* __restrict__ C,
                                                        const float* __restrict__ B) {
    __shared__ float As[16 * AS_STRIDE];
    const int tid  = threadIdx.x;
    const int lane = tid & 31;
    const int wave = tid >> 5;             // 0..7
    const int m0   = blockIdx.x * 16;

    // stage 16x1024 fp32 rows (64KB) into LDS, float2 granules, coalesced
    for (int f = tid; f < 16 * 512; f += 256) {
        const int r  = f >> 9;             // row 0..15
        const int cw = (f & 511) << 1;     // col 0..1022 step 2
        *(v2f*)(&As[r * AS_STRIDE + cw]) = *(const v2f*)(C + (m0 + r) * HD + cw);
    }
    __syncthreads();

    const int col   = lane & 15;
    const int koff  = 2 * (lane >> 4);
    const int rbase = 8 * (lane >> 4);
    const float* Asrow = &As[col * AS_STRIDE];
#pragma unroll 1
    for (int j = 0; j < 8; ++j) {
        const int n0 = (wave + 8 * j) * 16;     // 64 n-tiles / 8 waves
        v8f acc = {};
#pragma unroll 4
        for (int k = 0; k < HD; k += 4) {
            const int kk = k + koff;
            v2f a = *(const v2f*)(Asrow + kk);
            const float* Bp = B + kk * HD + n0 + col;
            v2f b = {Bp[0], Bp[HD]};
            acc = wmma_f32(a, b, acc);
        }
#pragma unroll
        for (int r = 0; r < 8; ++r)
            C[(m0 + rbase + r) * HD + n0 + col] = acc[r];   // safe: rows owned
    }
}

// ---------------------------------------------------------------------------
extern "C" void kernel_launch(void* const* d_in, const int* in_sizes, int n_in,
                              void* d_out, int out_size, void* d_ws, size_t ws_size,
                              hipStream_t stream) {
    const float* x        = (const float*)d_in[0];  // [512,64,1024]
    const float* w_hidden = (const float*)d_in[1];  // [1024,1024]
    const float* w_input  = (const float*)d_in[2];  // [1024,1024]
    const float* w_output = (const float*)d_in[3];  // [1024,1024]
    float* out = (float*)d_out;                     // outputs[512,64,1024] ++ h_last[64,1024]
    unsigned* bar = (unsigned*)d_ws;                // 2 words of barrier state

    // reset grid-barrier state (graph-capturable memset node)
    hipMemsetAsync(d_ws, 0, 64, stream);

    // Phase 1: XW = x @ w_input, written into d_out's outputs region
    xw_gemm<<<dim3(HD / 256, MTOT / 16), 128, 0, stream>>>(x, w_input, out);

    // Phase 2: sequential recurrence, XW slabs replaced by hidden states
    rnn_steps<<<64, 128, 0, stream>>>(w_hidden, out, bar);

    // h_last = h[511] (must copy before phase 3 overwrites it)
    hipMemcpyAsync(out + (size_t)LSTEPS * NH, out + (size_t)(LSTEPS - 1) * NH,
                   (size_t)NH * sizeof(float), hipMemcpyDeviceToDevice, stream);

    // Phase 3: outputs = H_ALL @ w_output, in place over d_out
    out_gemm_inplace<<<MTOT / 16, 256, 0, stream>>>(out, w_output);
}